// Classifier_82686710383323
// MI455X (gfx1250) — compile-verified
//
#include <hip/hip_runtime.h>
#include <hip/hip_bf16.h>

// ---------------------------------------------------------------------------
// SNN classifier for MI455X (gfx1250, wave32).
// Phase 1: fused GEMM chain (x@W0^T+b0 -> 7x 64x64 -> 64->10 head) using
//          V_WMMA_F32_16X16X4_F32; currents written once to workspace.
// Phase 2: LIF scan, store-bandwidth bound (~427 MB of output), mem kept in
//          registers, non-temporal 128-bit stores.
// ---------------------------------------------------------------------------

typedef float v2f __attribute__((ext_vector_type(2)));
typedef float v4f __attribute__((ext_vector_type(4)));
typedef float v8f __attribute__((ext_vector_type(8)));

#define NB     4096
#define NIN    784
#define NHID   64
#define NOUT   10
#define NLAY   8          // c0..c7
#define TSTEPS 25
#define BETA   0.95f

// output offsets (floats)
#define LH_ELEMS   ((size_t)NLAY * NB * NHID)          // 2,097,152
#define LO_ELEMS   ((size_t)NB * NOUT)                 // 40,960
#define SOFT_H_OFF ((size_t)TSTEPS * LH_ELEMS)         // 52,428,800
#define SPK_O_OFF  ((size_t)2 * TSTEPS * LH_ELEMS)     // 104,857,600
#define SOFT_O_OFF (SPK_O_OFF + (size_t)TSTEPS * LO_ELEMS)

static __device__ __forceinline__ v8f v8f_zero() {
    v8f z = {0.f, 0.f, 0.f, 0.f, 0.f, 0.f, 0.f, 0.f};
    return z;
}

#define WMMA_F32_16x16x4(a, b, c)                                            \
    __builtin_amdgcn_wmma_f32_16x16x4_f32(false, (a), false, (b), (short)0,  \
                                          (c), false, false)

// ---------------------------------------------------------------------------
// Phase 1: one wave (32 threads) per 16 batch rows; runs the full layer chain.
// LDS staging tile sc[16][66] (row stride 66 dwords -> bank-conflict-free).
// ---------------------------------------------------------------------------
__global__ __launch_bounds__(32) void snn_chain_kernel(
    const float* __restrict__ x,  const float* __restrict__ W0,
    const float* __restrict__ b0, const float* __restrict__ Wh,
    const float* __restrict__ bh, const float* __restrict__ Wo,
    const float* __restrict__ bo, float* __restrict__ cur_h,
    float* __restrict__ cur_o)
{
    __shared__ float sc[16 * 66];

    const int lane = threadIdx.x;        // 0..31
    const int half = lane >> 4;          // 0|1 (K sub-slice)
    const int lm   = lane & 15;          // row (A) / col (B,C)
    const int row0 = blockIdx.x * 16;    // 16 batch rows per wave

    // ------------------- layer 0: c0 = x @ W0^T + b0 (K = 784) -------------
    v8f acc[4];
#pragma unroll
    for (int j = 0; j < 4; ++j) acc[j] = v8f_zero();

    {
        const float* xr = x + (size_t)(row0 + lm) * NIN + 2 * half;
        for (int k = 0; k < NIN; k += 4) {
            v2f a = *(const v2f*)(xr + k);
#pragma unroll
            for (int j = 0; j < 4; ++j) {
                v2f b = *(const v2f*)(W0 + (size_t)(16 * j + lm) * NIN +
                                      2 * half + k);
                acc[j] = WMMA_F32_16x16x4(a, b, acc[j]);
            }
        }
#pragma unroll
        for (int j = 0; j < 4; ++j) acc[j] += b0[16 * j + lm];
    }

    // spill c0 to ws + LDS
#pragma unroll
    for (int j = 0; j < 4; ++j)
#pragma unroll
        for (int v = 0; v < 8; ++v) {
            const int r = v + 8 * half, c = 16 * j + lm;
            const float val = acc[j][v];
            cur_h[(size_t)(row0 + r) * NHID + c] = val;   // layer 0 slab
            sc[r * 66 + c] = val;
        }

    // ------------------- hidden chain: c_{i+1} = c_i @ Wh[i]^T + bh[i] -----
    for (int layer = 0; layer < NLAY - 1; ++layer) {
        __syncthreads();                                   // sc visible
        const float* W = Wh + (size_t)layer * NHID * NHID;

        v8f nacc[4];
#pragma unroll
        for (int j = 0; j < 4; ++j) nacc[j] = v8f_zero();

        for (int k = 0; k < NHID; k += 4) {
            v2f a = *(const v2f*)(&sc[lm * 66 + 2 * half + k]);
#pragma unroll
            for (int j = 0; j < 4; ++j) {
                v2f b = *(const v2f*)(W + (size_t)(16 * j + lm) * NHID +
                                      2 * half + k);
                nacc[j] = WMMA_F32_16x16x4(a, b, nacc[j]);
            }
        }
#pragma unroll
        for (int j = 0; j < 4; ++j) nacc[j] += bh[layer * NHID + 16 * j + lm];

        __syncthreads();                                   // WAR on sc
        float* dst = cur_h + (size_t)(layer + 1) * NB * NHID;
#pragma unroll
        for (int j = 0; j < 4; ++j)
#pragma unroll
            for (int v = 0; v < 8; ++v) {
                const int r = v + 8 * half, c = 16 * j + lm;
                const float val = nacc[j][v];
                dst[(size_t)(row0 + r) * NHID + c] = val;
                sc[r * 66 + c] = val;
            }
    }

    // ------------------- head: cur_o = c7 @ Wo^T + bo (cols 0..9) ----------
    __syncthreads();
    {
        const float msk = (lm < NOUT) ? 1.0f : 0.0f;
        const int   nc  = (lm < NOUT) ? lm : 0;            // clamped row of Wo
        v8f oacc = v8f_zero();
        for (int k = 0; k < NHID; k += 4) {
            v2f a = *(const v2f*)(&sc[lm * 66 + 2 * half + k]);
            v2f b = *(const v2f*)(Wo + (size_t)nc * NHID + 2 * half + k);
            b *= msk;                                      // no EXEC change
            oacc = WMMA_F32_16x16x4(a, b, oacc);
        }
        if (lm < NOUT) {
            const float bb = bo[nc];
#pragma unroll
            for (int v = 0; v < 8; ++v) {
                const int r = v + 8 * half;
                cur_o[(size_t)(row0 + r) * NOUT + lm] = oacc[v] + bb;
            }
        }
    }
}

// ---------------------------------------------------------------------------
// Phase 2: LIF dynamics. mem in registers, loop over T, NT stores.
// ---------------------------------------------------------------------------
static __device__ __forceinline__ void lif_step(float& m, float c, float& s,
                                                float& f) {
    m = BETA * m + c;
    s = (m > 1.0f) ? 1.0f : 0.0f;
    f = __builtin_amdgcn_rcpf(1.0f + __expf(1.0f - m));  // sigmoid(m - thr)
    m -= s;
}

__global__ __launch_bounds__(256) void lif_hidden_kernel(
    const float* __restrict__ cur, float* __restrict__ out_spk,
    float* __restrict__ out_soft)
{
    const size_t i = ((size_t)blockIdx.x * 256 + threadIdx.x) * 4;
    const v4f c4 = *(const v4f*)(cur + i);
    float cc[4] = {c4.x, c4.y, c4.z, c4.w};
    float m[4]  = {0.f, 0.f, 0.f, 0.f};

    for (int t = 0; t < TSTEPS; ++t) {
        float s[4], f[4];
#pragma unroll
        for (int e = 0; e < 4; ++e) lif_step(m[e], cc[e], s[e], f[e]);
        v4f sv = {s[0], s[1], s[2], s[3]};
        v4f fv = {f[0], f[1], f[2], f[3]};
        __builtin_nontemporal_store(sv, (v4f*)(out_spk  + (size_t)t * LH_ELEMS + i));
        __builtin_nontemporal_store(fv, (v4f*)(out_soft + (size_t)t * LH_ELEMS + i));
    }
}

__global__ __launch_bounds__(256) void lif_out_kernel(
    const float* __restrict__ cur, float* __restrict__ out_spk,
    float* __restrict__ out_soft)
{
    const size_t i = (size_t)blockIdx.x * 256 + threadIdx.x;
    const float c = cur[i];
    float m = 0.f;
    for (int t = 0; t < TSTEPS; ++t) {
        float s, f;
        lif_step(m, c, s, f);
        __builtin_nontemporal_store(s, out_spk  + (size_t)t * LO_ELEMS + i);
        __builtin_nontemporal_store(f, out_soft + (size_t)t * LO_ELEMS + i);
    }
}

// ---------------------------------------------------------------------------
extern "C" void kernel_launch(void* const* d_in, const int* in_sizes, int n_in,
                              void* d_out, int out_size, void* d_ws,
                              size_t ws_size, hipStream_t stream)
{
    const float* x  = (const float*)d_in[0];
    const float* W0 = (const float*)d_in[1];
    const float* b0 = (const float*)d_in[2];
    const float* Wh = (const float*)d_in[3];
    const float* bh = (const float*)d_in[4];
    const float* Wo = (const float*)d_in[5];
    const float* bo = (const float*)d_in[6];
    float* out = (float*)d_out;

    float* ws_h = (float*)d_ws;          // [8, 4096, 64] currents (8 MB)
    float* ws_o = ws_h + LH_ELEMS;       // [4096, 10] head currents (160 KB)

    // Phase 1: 256 one-wave workgroups, 16 batch rows each.
    snn_chain_kernel<<<NB / 16, 32, 0, stream>>>(x, W0, b0, Wh, bh, Wo, bo,
                                                 ws_h, ws_o);

    // Phase 2: store-bound LIF scans.
    lif_hidden_kernel<<<(unsigned)(LH_ELEMS / 4 / 256), 256, 0, stream>>>(
        ws_h, out, out + SOFT_H_OFF);
    lif_out_kernel<<<(unsigned)(LO_ELEMS / 256), 256, 0, stream>>>(
        ws_o, out + SPK_O_OFF, out + SOFT_O_OFF);
}